// DecodeDetections_75883482186161
// MI455X (gfx1250) — compile-verified
//
#include <hip/hip_runtime.h>
#include <stdint.h>

// Problem constants (from reference)
#define N_BOXES   24564
#define ROW       93          // N_CLASSES + 12
#define NCONF     80          // classes 1..80
#define TOPK      200
#define NBATCH    32
#define IMG_WH    512.0f
#define CONF_TH   0.5f

#define K1_ROWS   128         // rows (boxes) staged per block via TDM
#define K2_THREADS 1024
#define K2_PER    24          // ceil(24564/1024)

typedef unsigned int v4u __attribute__((ext_vector_type(4)));
typedef int          v8i __attribute__((ext_vector_type(8)));
typedef int          v4i __attribute__((ext_vector_type(4)));

// ---- monotone order-preserving float<->u32 key mapping ----
__device__ __forceinline__ unsigned int ford(float f) {
  unsigned int u = __float_as_uint(f);
  return u ^ ((u >> 31) ? 0xFFFFFFFFu : 0x80000000u);
}
__device__ __forceinline__ float iord(unsigned int o) {
  unsigned int u = (o & 0x80000000u) ? (o ^ 0x80000000u) : ~o;
  return __uint_as_float(u);
}

// 64-bit xor-shuffle built from two 32-bit wave32 shuffles
__device__ __forceinline__ unsigned long long shflx64(unsigned long long v, int off) {
  unsigned int lo = (unsigned int)v;
  unsigned int hi = (unsigned int)(v >> 32);
  lo = __shfl_xor(lo, off, 32);
  hi = __shfl_xor(hi, off, 32);
  return ((unsigned long long)hi << 32) | lo;
}
__device__ __forceinline__ unsigned long long wave_max_u64(unsigned long long v) {
#pragma unroll
  for (int off = 16; off > 0; off >>= 1) {
    unsigned long long o = shflx64(v, off);
    v = (o > v) ? o : v;
  }
  return v;
}

// ---------------------------------------------------------------------------
// Kernel 1: batch-0 max_conf per box -> 64-bit sort keys.
// TDM stages a 128x93 f32 tile into LDS (one DMA per workgroup), then each
// thread reduces its row from LDS (stride 93 = odd -> bank-conflict-free).
// ---------------------------------------------------------------------------
__global__ __launch_bounds__(K1_ROWS) void k1_scores(
    const float* __restrict__ yp, unsigned long long* __restrict__ keys) {
  __shared__ float tile[K1_ROWS * ROW];   // 47,616 bytes
  int start = blockIdx.x * K1_ROWS;
  if (start > N_BOXES - K1_ROWS) start = N_BOXES - K1_ROWS;  // idempotent overlap
  const int t = threadIdx.x;

#if __has_builtin(__builtin_amdgcn_tensor_load_to_lds) && __has_builtin(__builtin_amdgcn_s_wait_tensorcnt)
  if (t < 32) {   // one wave issues the TDM op (EXEC-independent; branch skips others)
    unsigned long long ga = (unsigned long long)(uintptr_t)(yp + (size_t)start * ROW);
    unsigned int lds_off = (unsigned int)(uintptr_t)(&tile[0]);   // low 32 bits = LDS offset
    // D# group 0: count=1 | lds_addr | global_addr[56:0] | type=2
    v4u g0;
    g0.x = 1u;
    g0.y = lds_off;
    g0.z = (unsigned int)ga;
    g0.w = (unsigned int)((ga >> 32) & 0x1FFFFFFull) | (2u << 30);
    // D# group 1: data_size=4B, 2-D tile 93 x 128, tensor dims == tile dims,
    // dim0 stride = 93 elements. (Bit layout per CDNA5 ISA.)
    v8i g1;
    g1[0] = (int)(2u << 16);                    // workgroup_mask=0, data_size=2 (4B)
    g1[1] = (int)((unsigned)ROW << 16);         // tensor_dim0[15:0] = 93
    g1[2] = (int)((unsigned)K1_ROWS << 16);     // tensor_dim0[31:16]=0 | tensor_dim1[15:0]=128
    g1[3] = (int)((unsigned)ROW << 16);         // tensor_dim1[31:16]=0 | tile_dim0=93
    g1[4] = (int)(unsigned)K1_ROWS;             // tile_dim1=128 | tile_dim2=0
    g1[5] = ROW;                                // tensor_dim0_stride[31:0] = 93
    g1[6] = 0;                                  // stride hi | tensor_dim1_stride lo (unused, 2D)
    g1[7] = 0;
    v4i gz4 = {0, 0, 0, 0};                     // groups 2/3: tile_dim3/4 = 0 -> unused
    v8i gz8 = {0, 0, 0, 0, 0, 0, 0, 0};         // extra group (clang-23 6-arg form): zeros
    __builtin_amdgcn_tensor_load_to_lds(g0, g1, gz4, gz4, gz8, 0);
    __builtin_amdgcn_s_wait_tensorcnt(0);
  }
  __syncthreads();
#else
  for (int i = t; i < K1_ROWS * ROW; i += K1_ROWS)
    tile[i] = yp[(size_t)start * ROW + i];
  __syncthreads();
#endif

  // max over class channels 1..80 of this thread's row
  float m = tile[t * ROW + 1];
#pragma unroll
  for (int c = 2; c <= NCONF; ++c) {
    float v = tile[t * ROW + c];
    m = (v > m) ? v : m;
  }
  const unsigned int r = (unsigned int)(start + t);
  keys[r] = ((unsigned long long)ford(m) << 32) | (unsigned int)(~r);
}

// ---------------------------------------------------------------------------
// Kernel 2: exact top-200 of 24564 keys, single 32-wave workgroup.
// Each thread keeps its 24 keys in registers; winner removal is register-local
// (no cross-wave memory coherence needed). Tie-break matches jax.lax.top_k.
// ---------------------------------------------------------------------------
__global__ __launch_bounds__(K2_THREADS) void k2_select(
    const unsigned long long* __restrict__ keys, unsigned int* __restrict__ topIdx) {
  __shared__ unsigned long long part[32];
  __shared__ unsigned long long swin;
  const int t = threadIdx.x;
  const int lane = t & 31;
  const int wid = t >> 5;

  unsigned long long lk[K2_PER];
#pragma unroll
  for (int j = 0; j < K2_PER; ++j) {
    int i = t + j * K2_THREADS;
    lk[j] = (i < N_BOXES) ? keys[i] : 0ull;
  }

  for (int k = 0; k < TOPK; ++k) {
    unsigned long long m = 0ull;
#pragma unroll
    for (int j = 0; j < K2_PER; ++j) m = (lk[j] > m) ? lk[j] : m;
    m = wave_max_u64(m);
    if (lane == 0) part[wid] = m;
    __syncthreads();
    if (wid == 0) {
      unsigned long long v = part[lane];
      v = wave_max_u64(v);
      if (lane == 0) {
        swin = v;
        topIdx[k] = ~(unsigned int)v;     // recover box index from key low bits
      }
    }
    __syncthreads();
    const unsigned int w = ~(unsigned int)swin;
    if ((int)(w % K2_THREADS) == t) {
      const int slot = (int)(w / K2_THREADS);
#pragma unroll
      for (int j = 0; j < K2_PER; ++j)
        if (j == slot) lk[j] = 0ull;      // remove winner (register-local)
    }
  }
}

// ---------------------------------------------------------------------------
// Kernel 3: gather + decode. One wave32 per output row (b, k):
// lanes reduce 80 class scores (argmax = first max), lane 0 decodes the box.
// ---------------------------------------------------------------------------
__global__ __launch_bounds__(256) void k3_decode(
    const float* __restrict__ yp, const unsigned int* __restrict__ topIdx,
    float* __restrict__ out) {
  const int lane = threadIdx.x & 31;
  const int g = (int)((blockIdx.x * blockDim.x + threadIdx.x) >> 5);   // global wave id
  if (g >= NBATCH * TOPK) return;
  const int b = g / TOPK;
  const int k = g - b * TOPK;
  const unsigned int idx = topIdx[k];
  const float* row = yp + ((size_t)b * N_BOXES + idx) * ROW;

  // classes c in 0..79 live at channels 1..80; key tie-breaks to lowest class
  float v0 = row[1 + lane];
  float v1 = row[1 + lane + 32];
  unsigned long long key =
      ((unsigned long long)ford(v0) << 32) | (unsigned int)(~(unsigned int)lane);
  unsigned long long kb =
      ((unsigned long long)ford(v1) << 32) | (unsigned int)(~(unsigned int)(lane + 32));
  if (kb > key) key = kb;
  if (lane < (NCONF - 64)) {
    float v2 = row[1 + lane + 64];
    unsigned long long kc =
        ((unsigned long long)ford(v2) << 32) | (unsigned int)(~(unsigned int)(lane + 64));
    if (kc > key) key = kc;
  }
  key = wave_max_u64(key);

  if (lane == 0) {
    const float maxc = iord((unsigned int)(key >> 32));
    const unsigned int cls = ~(unsigned int)key;
    const float a81 = row[81], a82 = row[82], a83 = row[83], a84 = row[84];
    const float a85 = row[85], a86 = row[86], a87 = row[87], a88 = row[88];
    const float a89 = row[89], a90 = row[90], a91 = row[91], a92 = row[92];
    const float cx = a81 * a89 * a87 + a85;
    const float cy = a82 * a90 * a88 + a86;
    const float w  = expf(a83 * a91) * a87;
    const float h  = expf(a84 * a92) * a88;
    float* o = out + (size_t)g * 7;
    o[0] = (maxc > CONF_TH) ? 1.0f : 0.0f;
    o[1] = (float)cls;
    o[2] = maxc;
    o[3] = (cx - 0.5f * w) * IMG_WH;
    o[4] = (cy - 0.5f * h) * IMG_WH;
    o[5] = (cx + 0.5f * w) * IMG_WH;
    o[6] = (cy + 0.5f * h) * IMG_WH;
  }
}

// ---------------------------------------------------------------------------
extern "C" void kernel_launch(void* const* d_in, const int* in_sizes, int n_in,
                              void* d_out, int out_size, void* d_ws, size_t ws_size,
                              hipStream_t stream) {
  const float* yp = (const float*)d_in[0];
  float* out = (float*)d_out;
  unsigned long long* keys = (unsigned long long*)d_ws;                      // 24564 * 8 B
  unsigned int* topIdx = (unsigned int*)((char*)d_ws + (size_t)N_BOXES * 8); // 200 * 4 B

  const int k1_blocks = (N_BOXES + K1_ROWS - 1) / K1_ROWS;  // 192
  k1_scores<<<k1_blocks, K1_ROWS, 0, stream>>>(yp, keys);
  k2_select<<<1, K2_THREADS, 0, stream>>>(keys, topIdx);
  const int k3_waves = NBATCH * TOPK;                        // 6400
  k3_decode<<<(k3_waves * 32) / 256, 256, 0, stream>>>(yp, topIdx, out);
}